// GFMLayer_90460601189050
// MI455X (gfx1250) — compile-verified
//
#include <hip/hip_runtime.h>
#include <hip/hip_bf16.h>

typedef __attribute__((ext_vector_type(2))) float v2f;
typedef __attribute__((ext_vector_type(8))) float v8f;

#define NNODE_H 128
#define NHEAD 8
#define HDIM 16

static __device__ inline float atomAddF(float* p, float v) {
    return __hip_atomic_fetch_add(p, v, __ATOMIC_RELAXED, __HIP_MEMORY_SCOPE_AGENT);
}

// D = A(16x4,f32) x B(4x16,f32) + C(16x16,f32), wave32.
// Fragment layout per ISA 7.12.2:
//  A: lane l(0-15): vgpr0=A[l, 2*hi], vgpr1=A[l, 2*hi+1]   (hi = lane>>4)
//  B: lane l: vgpr0=B[2*hi, l], vgpr1=B[2*hi+1, l]
//  C/D: vgpr i = C[i + 8*hi, l]
static __device__ inline v8f wmma_f32_16x16x4(v2f a, v2f b, v8f c) {
#if __has_builtin(__builtin_amdgcn_wmma_f32_16x16x4_f32)
    return __builtin_amdgcn_wmma_f32_16x16x4_f32(false, a, false, b, (short)0, c,
                                                 false, false);
#else
    // Exact cross-lane emulation fallback (only if builtin is absent).
    int lane = threadIdx.x & 31;
    int l = lane & 15, hi = lane >> 4;
    float b0 = __shfl(a.x, 0, 32); (void)b0;
    float bx0 = __shfl(b.x, l, 32);
    float bx1 = __shfl(b.y, l, 32);
    float bx2 = __shfl(b.x, l + 16, 32);
    float bx3 = __shfl(b.y, l + 16, 32);
    #pragma unroll
    for (int i = 0; i < 8; ++i) {
        int m = i + 8 * hi;
        float a0 = __shfl(a.x, m, 32);
        float a1 = __shfl(a.y, m, 32);
        float a2 = __shfl(a.x, m + 16, 32);
        float a3 = __shfl(a.y, m + 16, 32);
        c[i] = fmaf(a0, bx0, fmaf(a1, bx1, fmaf(a2, bx2, fmaf(a3, bx3, c[i]))));
    }
    return c;
#endif
}

// ---------------------------------------------------------------------------
// cq[n] = bq[n] + sum_h qemb[h]*Wq[n,128+h];  same for ck. One block, 128 thr.
__global__ void cqck_kernel(const float* __restrict__ Wq, const float* __restrict__ bq,
                            const float* __restrict__ Wk, const float* __restrict__ bk,
                            const float* __restrict__ qemb,
                            float* __restrict__ cq, float* __restrict__ ck) {
    int n = threadIdx.x;
    float s1 = bq[n], s2 = bk[n];
    const float* wq = Wq + (size_t)n * 256 + 128;
    const float* wk = Wk + (size_t)n * 256 + 128;
    #pragma unroll 8
    for (int h = 0; h < 128; ++h) {
        float q = qemb[h];
        s1 = fmaf(q, wq[h], s1);
        s2 = fmaf(q, wk[h], s2);
    }
    cq[n] = s1;
    ck[n] = s2;
}

// ---------------------------------------------------------------------------
// out[N,128] = A[N,128] @ W[:, :128]^T + bias  (W row-major [128, wld])
// 256 threads = 8 waves; each block does 16 output rows x 128 cols.
#define LDA_LDS 132
__global__ void gemm128_kernel(const float* __restrict__ A, const float* __restrict__ W,
                               int wld, const float* __restrict__ bias,
                               float* __restrict__ out) {
    __shared__ float ldsX[16 * LDA_LDS];
    int tid = threadIdx.x;
    int m0 = blockIdx.x * 16;
    // stage 16x128 A-tile into LDS (padded rows: conflict-free K-reads)
    for (int idx = tid; idx < 16 * 128; idx += 256) {
        int r = idx >> 7, c = idx & 127;
        ldsX[r * LDA_LDS + c] = A[(size_t)(m0 + r) * 128 + c];
    }
    __syncthreads();

    int wave = tid >> 5;
    int lane = tid & 31;
    int l = lane & 15, hi = lane >> 4;
    int n0 = wave * 16;

    v8f acc = {0.f, 0.f, 0.f, 0.f, 0.f, 0.f, 0.f, 0.f};
    const float* wrow = W + (size_t)(n0 + l) * wld + 2 * hi;
    const float* arow = ldsX + l * LDA_LDS + 2 * hi;
    #pragma unroll 8
    for (int k = 0; k < 128; k += 4) {
        v2f a = {arow[k], arow[k + 1]};
        v2f b = {wrow[k], wrow[k + 1]};
        acc = wmma_f32_16x16x4(a, b, acc);
    }
    float bb = bias[n0 + l];
    #pragma unroll
    for (int i = 0; i < 8; ++i) {
        int row = m0 + i + 8 * hi;
        out[(size_t)row * 128 + n0 + l] = acc[i] + bb;
    }
}

// ---------------------------------------------------------------------------
// Fused FFN: F = gelu(X1 @ W1^T + b1) @ W2^T + b2 ; 16 rows per block.
// H1 tile (16x512) lives entirely in LDS (no 205 MB global intermediate).
#define LDH_LDS 516
__global__ void ffn_kernel(const float* __restrict__ X1, const float* __restrict__ W1,
                           const float* __restrict__ b1, const float* __restrict__ W2,
                           const float* __restrict__ b2, float* __restrict__ out) {
    __shared__ float ldsX[16 * LDA_LDS];
    __shared__ float ldsH[16 * LDH_LDS];
    int tid = threadIdx.x;
    int m0 = blockIdx.x * 16;
    for (int idx = tid; idx < 16 * 128; idx += 256) {
        int r = idx >> 7, c = idx & 127;
        ldsX[r * LDA_LDS + c] = X1[(size_t)(m0 + r) * 128 + c];
    }
    __syncthreads();

    int wave = tid >> 5;
    int lane = tid & 31;
    int l = lane & 15, hi = lane >> 4;

    // stage 1: H1 = gelu(X1 @ W1^T + b1), 32 n-tiles over 8 waves
    for (int t = wave; t < 32; t += 8) {
        int n0 = t * 16;
        v8f acc = {0.f, 0.f, 0.f, 0.f, 0.f, 0.f, 0.f, 0.f};
        const float* wrow = W1 + (size_t)(n0 + l) * 128 + 2 * hi;
        const float* arow = ldsX + l * LDA_LDS + 2 * hi;
        #pragma unroll 8
        for (int k = 0; k < 128; k += 4) {
            v2f a = {arow[k], arow[k + 1]};
            v2f b = {wrow[k], wrow[k + 1]};
            acc = wmma_f32_16x16x4(a, b, acc);
        }
        float bb = b1[n0 + l];
        #pragma unroll
        for (int i = 0; i < 8; ++i) {
            float x = acc[i] + bb;
            float g = 0.5f * x * (1.0f + erff(x * 0.70710678118654752440f));
            ldsH[(i + 8 * hi) * LDH_LDS + n0 + l] = g;
        }
    }
    __syncthreads();

    // stage 2: F = H1 @ W2^T + b2 (K=512), 8 n-tiles, one per wave
    {
        int n0 = wave * 16;
        v8f acc = {0.f, 0.f, 0.f, 0.f, 0.f, 0.f, 0.f, 0.f};
        const float* wrow = W2 + (size_t)(n0 + l) * 512 + 2 * hi;
        const float* arow = ldsH + l * LDH_LDS + 2 * hi;
        #pragma unroll 4
        for (int k = 0; k < 512; k += 4) {
            v2f a = {arow[k], arow[k + 1]};
            v2f b = {wrow[k], wrow[k + 1]};
            acc = wmma_f32_16x16x4(a, b, acc);
        }
        float bb = b2[n0 + l];
        #pragma unroll
        for (int i = 0; i < 8; ++i) {
            int row = m0 + i + 8 * hi;
            out[(size_t)row * 128 + n0 + l] = acc[i] + bb;
        }
    }
}

// ---------------------------------------------------------------------------
// LayerNorm: out[r,:] = LN(a[r,:] + b[r,:]) * g + be. One block (128 thr) per row.
__global__ void ln_kernel(const float* __restrict__ a, const float* __restrict__ b,
                          const float* __restrict__ g, const float* __restrict__ be,
                          float* __restrict__ out) {
    __shared__ float red[128];
    int r = blockIdx.x, t = threadIdx.x;
    float x = a[(size_t)r * 128 + t] + b[(size_t)r * 128 + t];
    red[t] = x;
    __syncthreads();
    for (int s = 64; s > 0; s >>= 1) {
        if (t < s) red[t] += red[t + s];
        __syncthreads();
    }
    float mu = red[0] * (1.0f / 128.0f);
    __syncthreads();
    float d = x - mu;
    red[t] = d * d;
    __syncthreads();
    for (int s = 64; s > 0; s >>= 1) {
        if (t < s) red[t] += red[t + s];
        __syncthreads();
    }
    float var = red[0] * (1.0f / 128.0f);
    out[(size_t)r * 128 + t] = d * __frsqrt_rn(var + 1e-5f) * g[t] + be[t];
}

// ---------------------------------------------------------------------------
// Edge pass 1: scores + atomic segment-max (exact uint-trick; init 0 == max(0,.))
__global__ void edge_score_kernel(const float* __restrict__ Q, const float* __restrict__ K,
                                  const float* __restrict__ rel, const int* __restrict__ ei,
                                  const int* __restrict__ etype, float* __restrict__ SB,
                                  float* __restrict__ MB, long long E) {
    long long idx = (long long)blockIdx.x * 256 + threadIdx.x;
    if (idx >= E * NHEAD) return;
    long long e = idx >> 3;
    int h = (int)(idx & 7);
    int src = ei[e], dst = ei[E + e], ty = etype[e];
    const float* qp = Q + (size_t)dst * 128 + h * 16;
    const float* kp = K + (size_t)src * 128 + h * 16;
    const float* rp = rel + (size_t)ty * 128 + h * 16;
    float s = 0.f;
    #pragma unroll
    for (int d = 0; d < 16; ++d) s = fmaf(qp[d], kp[d] + rp[d], s);
    s *= 0.25f;  // 1/sqrt(16)
    SB[idx] = s;
    if (s > 0.f)
        atomicMax((unsigned int*)(MB + (size_t)dst * NHEAD + h), __float_as_uint(s));
}

// Edge pass 2: exp(s - max) in place + denominator segment-sum
__global__ void edge_exp_kernel(float* __restrict__ SB, const float* __restrict__ MB,
                                float* __restrict__ DB, const int* __restrict__ ei,
                                long long E) {
    long long idx = (long long)blockIdx.x * 256 + threadIdx.x;
    if (idx >= E * NHEAD) return;
    long long e = idx >> 3;
    int h = (int)(idx & 7);
    int dst = ei[E + e];
    float es = __expf(SB[idx] - MB[(size_t)dst * NHEAD + h]);
    es = expf(SB[idx] - MB[(size_t)dst * NHEAD + h]);  // full-precision exp
    SB[idx] = es;
    atomAddF(DB + (size_t)dst * NHEAD + h, es);
}

// Edge pass 3: agg[dst] += w * v[src]
__global__ void edge_agg_kernel(const float* __restrict__ SB, const float* __restrict__ DB,
                                const float* __restrict__ V, const int* __restrict__ ei,
                                float* __restrict__ AB, long long E) {
    long long idx = (long long)blockIdx.x * 256 + threadIdx.x;
    if (idx >= E * NHEAD) return;
    long long e = idx >> 3;
    int h = (int)(idx & 7);
    int src = ei[e], dst = ei[E + e];
    float w = SB[idx] / (DB[(size_t)dst * NHEAD + h] + 1e-8f);
    const float* vp = V + (size_t)src * 128 + h * 16;
    float* ap = AB + (size_t)dst * 128 + h * 16;
    #pragma unroll
    for (int d = 0; d < 16; ++d) atomAddF(ap + d, w * vp[d]);
}

// ---------------------------------------------------------------------------
// Relational block (R=100, tiny): t[r,k,o] = sum_h relemb[r,h]*relW[k,o,h] + relb[k,o]
__global__ void rel_t_kernel(const float* __restrict__ relemb, const float* __restrict__ relW,
                             const float* __restrict__ relb, float* __restrict__ combined,
                             int R) {
    int idx = blockIdx.x * 256 + threadIdx.x;
    if (idx >= R * 512) return;
    int r = idx / 512, ko = idx % 512;
    int k = ko >> 7, o = ko & 127;
    const float* er = relemb + (size_t)r * 128;
    const float* wr = relW + ((size_t)k * 128 + o) * 128;
    float s = relb[k * 128 + o];
    #pragma unroll 8
    for (int h = 0; h < 128; ++h) s = fmaf(er[h], wr[h], s);
    combined[idx] = s;
}

// relc[r,o] = bc[o] + combined[r,:] . Wc[o,:]
__global__ void rel_c_kernel(const float* __restrict__ combined, const float* __restrict__ Wc,
                             const float* __restrict__ bc, float* __restrict__ relc, int R) {
    int idx = blockIdx.x * 256 + threadIdx.x;
    if (idx >= R * 128) return;
    int r = idx >> 7, o = idx & 127;
    const float* cr = combined + (size_t)r * 512;
    const float* wr = Wc + (size_t)o * 512;
    float s = bc[o];
    #pragma unroll 8
    for (int j = 0; j < 512; ++j) s = fmaf(cr[j], wr[j], s);
    relc[idx] = s;
}

// ---------------------------------------------------------------------------
extern "C" void kernel_launch(void* const* d_in, const int* in_sizes, int n_in,
                              void* d_out, int out_size, void* d_ws, size_t ws_size,
                              hipStream_t stream) {
    const float* X     = (const float*)d_in[0];
    const float* qemb  = (const float*)d_in[1];
    const int*   ei    = (const int*)d_in[2];
    const int*   etype = (const int*)d_in[3];
    const float* relemb= (const float*)d_in[4];
    const float* Wq = (const float*)d_in[5];  const float* bq = (const float*)d_in[6];
    const float* Wk = (const float*)d_in[7];  const float* bk = (const float*)d_in[8];
    const float* Wv = (const float*)d_in[9];  const float* bv = (const float*)d_in[10];
    const float* Wo = (const float*)d_in[11]; const float* bo = (const float*)d_in[12];
    const float* n1g = (const float*)d_in[13]; const float* n1b = (const float*)d_in[14];
    const float* n2g = (const float*)d_in[15]; const float* n2b = (const float*)d_in[16];
    const float* W1 = (const float*)d_in[17]; const float* b1 = (const float*)d_in[18];
    const float* W2 = (const float*)d_in[19]; const float* b2 = (const float*)d_in[20];
    const float* relW = (const float*)d_in[21]; const float* relb = (const float*)d_in[22];
    const float* Wc = (const float*)d_in[23]; const float* bc = (const float*)d_in[24];
    const float* rng_ = (const float*)d_in[25]; const float* rnb = (const float*)d_in[26];

    const long long N = in_sizes[0] / 128;
    const long long E = in_sizes[3];
    const int R = in_sizes[4] / 128;

    float* ws = (float*)d_ws;
    const size_t NHf = (size_t)N * 128;
    float* QB = ws;                       // Q, later attn_out
    float* KB = QB + NHf;                 // K, later X1
    float* VB = KB + NHf;                 // V
    float* SB = VB + NHf;                 // scores/exp (E*8), later F
    float* MB = SB + (size_t)E * NHEAD;   // seg max  (N*8)
    float* DB = MB + (size_t)N * NHEAD;   // denom    (N*8)
    float* AB = DB + (size_t)N * NHEAD;   // agg      (N*128)
    float* cq = AB + NHf;
    float* ck = cq + 128;
    float* combined = ck + 128;           // R*512
    float* relc = combined + (size_t)R * 512;

    float* OUTX = (float*)d_out;
    float* OUTR = OUTX + NHf;

    const int mblocks = (int)(N / 16);                       // N % 16 == 0
    const int eblocks = (int)((E * NHEAD + 255) / 256);

    // folded query biases
    cqck_kernel<<<1, 128, 0, stream>>>(Wq, bq, Wk, bk, qemb, cq, ck);

    // Q/K/V projections (WMMA fp32)
    gemm128_kernel<<<mblocks, 256, 0, stream>>>(X, Wq, 256, cq, QB);
    gemm128_kernel<<<mblocks, 256, 0, stream>>>(X, Wk, 256, ck, KB);
    gemm128_kernel<<<mblocks, 256, 0, stream>>>(X, Wv, 128, bv, VB);

    // zero max/denom/agg (contiguous region)
    hipMemsetAsync(MB, 0, (size_t)(N * NHEAD * 2 + N * 128) * sizeof(float), stream);

    // sparse attention
    edge_score_kernel<<<eblocks, 256, 0, stream>>>(QB, KB, relemb, ei, etype, SB, MB, E);
    edge_exp_kernel<<<eblocks, 256, 0, stream>>>(SB, MB, DB, ei, E);
    edge_agg_kernel<<<eblocks, 256, 0, stream>>>(SB, DB, VB, ei, AB, E);

    // output projection + LN1 (attn_out reuses QB, X1 reuses KB)
    gemm128_kernel<<<mblocks, 256, 0, stream>>>(AB, Wo, 128, bo, QB);
    ln_kernel<<<(int)N, 128, 0, stream>>>(X, QB, n1g, n1b, KB);

    // fused FFN (F reuses SB) + LN2 -> final node output
    ffn_kernel<<<mblocks, 256, 0, stream>>>(KB, W1, b1, W2, b2, SB);
    ln_kernel<<<(int)N, 128, 0, stream>>>(KB, SB, n2g, n2b, OUTX);

    // relational block
    rel_t_kernel<<<(R * 512 + 255) / 256, 256, 0, stream>>>(relemb, relW, relb, combined, R);
    rel_c_kernel<<<(R * 128 + 255) / 256, 256, 0, stream>>>(combined, Wc, bc, relc, R);
    ln_kernel<<<R, 128, 0, stream>>>(relemb, relc, rng_, rnb, OUTR);
}